// GraphConvNet_74629351735801
// MI455X (gfx1250) — compile-verified
//
#include <hip/hip_runtime.h>
#include <stdint.h>

typedef __attribute__((ext_vector_type(2))) float v2f;
typedef __attribute__((ext_vector_type(8))) float v8f;

#define N_NODES    100000
#define N_EDGES    3200000
#define NUM_GRAPHS 1024
#define MTILES     (N_NODES / 16)   // 6250, exact

// ---------------- degree / normalization ----------------

__global__ void init_deg_kernel(float* deg, int n) {
    int i = blockIdx.x * blockDim.x + threadIdx.x;
    if (i < n) deg[i] = 1.0f;   // self-loop contributes 1 to every node's degree
}

__global__ void accum_deg_kernel(const int* __restrict__ ei, float* deg, int E) {
    int e = blockIdx.x * blockDim.x + threadIdx.x;
    if (e < E) {
        __builtin_prefetch(&ei[E + e + 8192], 0, 0);
        atomicAdd(&deg[ei[E + e]], 1.0f);   // dst row of edge_index
    }
}

__global__ void deg_to_dinv_kernel(float* deg, int n) {
    int i = blockIdx.x * blockDim.x + threadIdx.x;
    if (i < n) {
        float d = deg[i];
        deg[i] = (d > 0.0f) ? rsqrtf(d) : 0.0f;
    }
}

// ---------------- WMMA GEMM: C[M,N] = (relu?)A[M,K] @ W[K,N] ----------------
// One wave32 computes one 16x16 output tile with V_WMMA_F32_16X16X4_F32.
// A-frag: lane holds row (lane&15), Ks {half*2, half*2+1}; half = lane>>4.
// B-frag: lane holds col (lane&15), same K pair.
// C-frag: VGPR v holds row (v + half*8), col (lane&15).

template <int K, int N, bool RELU_IN>
__global__ void __launch_bounds__(256)
gemm_wmma_kernel(const float* __restrict__ A, const float* __restrict__ W,
                 float* __restrict__ C, int mtiles) {
    const int ntiles = N / 16;
    int wave = blockIdx.x * (blockDim.x >> 5) + (threadIdx.x >> 5);
    if (wave >= mtiles * ntiles) return;         // wave-uniform: EXEC stays all-1s
    int lane = threadIdx.x & 31;
    int tm = wave / ntiles;
    int tn = wave - tm * ntiles;
    int l15  = lane & 15;
    int half = lane >> 4;

    const float* arow = A + (size_t)(tm * 16 + l15) * K + half * 2;
    const float* wcol = W + tn * 16 + l15;

    v8f acc = {};
#pragma unroll
    for (int kb = 0; kb < K; kb += 4) {
        float a0 = arow[kb];
        float a1 = arow[kb + 1];
        if (RELU_IN) { a0 = fmaxf(a0, 0.0f); a1 = fmaxf(a1, 0.0f); }
        v2f a; a.x = a0; a.y = a1;
        v2f b;
        b.x = wcol[(size_t)(kb + half * 2 + 0) * N];
        b.y = wcol[(size_t)(kb + half * 2 + 1) * N];
        acc = __builtin_amdgcn_wmma_f32_16x16x4_f32(
            /*neg_a=*/false, a, /*neg_b=*/false, b,
            /*c_mod=*/(short)0, acc, /*reuse_a=*/false, /*reuse_b=*/false);
    }

    float* crow = C + (size_t)(tm * 16 + half * 8) * N + tn * 16 + l15;
#pragma unroll
    for (int v = 0; v < 8; ++v) crow[(size_t)v * N] = acc[v];
}

// ---------------- aggregation ----------------
// out[i,f] = bias[f] + dinv[i]^2 * t[i,f]   (bias + self-loop folded in)

template <int F>
__global__ void init_out_kernel(const float* __restrict__ t, const float* __restrict__ dinv,
                                const float* __restrict__ bias, float* __restrict__ out, int n) {
    int idx = blockIdx.x * blockDim.x + threadIdx.x;
    if (idx >= n * F) return;
    int i = idx / F;          // F is a power of two -> shift
    int f = idx & (F - 1);
    float di = dinv[i];
    out[idx] = bias[f] + di * di * t[idx];
}

// out[dst] += dinv[src]*dinv[dst] * t[src]  (4 features per thread, blockIdx.y = chunk)
template <int F>
__global__ void scatter_edges_kernel(const float* __restrict__ t, const int* __restrict__ ei,
                                     const float* __restrict__ dinv, float* __restrict__ out,
                                     int E) {
    int e = blockIdx.x * blockDim.x + threadIdx.x;
    if (e >= E) return;
    __builtin_prefetch(&ei[e + 8192], 0, 0);
    int s = ei[e];
    int d = ei[E + e];
    float w = dinv[s] * dinv[d];
    int c4 = blockIdx.y * 4;
    float4 v = *(const float4*)(t + (size_t)s * F + c4);
    float* op = out + (size_t)d * F + c4;
    atomicAdd(op + 0, w * v.x);
    atomicAdd(op + 1, w * v.y);
    atomicAdd(op + 2, w * v.z);
    atomicAdd(op + 3, w * v.w);
}

// ---------------- pooling + MLP head ----------------

__global__ void zero_pool_kernel(float* sums, float* cnt) {
    int i = blockIdx.x * blockDim.x + threadIdx.x;
    if (i < NUM_GRAPHS * 64) sums[i] = 0.0f;
    if (i < NUM_GRAPHS) cnt[i] = 0.0f;
}

__global__ void pool_accum_kernel(const float* __restrict__ h3, const int* __restrict__ batch,
                                  float* sums, float* cnt, int n) {
    int idx = blockIdx.x * blockDim.x + threadIdx.x;
    if (idx >= n * 64) return;
    int i = idx >> 6;
    int f = idx & 63;
    int g = batch[i];
    atomicAdd(&sums[g * 64 + f], fmaxf(h3[idx], 0.0f));   // relu fused here
    if (f == 0) atomicAdd(&cnt[g], 1.0f);
}

__global__ void __launch_bounds__(64)
head_kernel(const float* __restrict__ sums, const float* __restrict__ cnt,
            const float* __restrict__ Wl1, const float* __restrict__ bl1,
            const float* __restrict__ Wl2, const float* __restrict__ bl2,
            float* __restrict__ out) {
    __shared__ float pooled[64];
    __shared__ float hid[64];
    int g = blockIdx.x;
    int f = threadIdx.x;
    float c = cnt[g];
    pooled[f] = sums[g * 64 + f] / fmaxf(c, 1.0f);
    __syncthreads();
    float acc = bl1[f];
#pragma unroll
    for (int k = 0; k < 64; ++k) acc += pooled[k] * Wl1[k * 64 + f];
    hid[f] = fmaxf(acc, 0.0f);
    __syncthreads();
    if (f < 2) {
        float o = bl2[f];
#pragma unroll
        for (int k = 0; k < 64; ++k) o += hid[k] * Wl2[k * 2 + f];
        out[g * 2 + f] = o;
    }
}

// ---------------- host-side launch ----------------

extern "C" void kernel_launch(void* const* d_in, const int* in_sizes, int n_in,
                              void* d_out, int out_size, void* d_ws, size_t ws_size,
                              hipStream_t stream) {
    const float* x     = (const float*)d_in[0];
    const int*   ei    = (const int*)d_in[1];     // [2, E]: row 0 = src, row 1 = dst
    const int*   batch = (const int*)d_in[2];
    const float* W1    = (const float*)d_in[3];
    const float* b1    = (const float*)d_in[4];
    const float* W2    = (const float*)d_in[5];
    const float* b2    = (const float*)d_in[6];
    const float* W3    = (const float*)d_in[7];
    const float* b3    = (const float*)d_in[8];
    const float* Wl1   = (const float*)d_in[9];
    const float* bl1   = (const float*)d_in[10];
    const float* Wl2   = (const float*)d_in[11];
    const float* bl2   = (const float*)d_in[12];
    float* out = (float*)d_out;

    // workspace carve-up (floats), 16B-aligned slices
    float* ws   = (float*)d_ws;
    float* dinv = ws;  ws += ((N_NODES + 255) / 256) * 256;       // deg -> dinv in place
    float* bufT = ws;  ws += (size_t)N_NODES * 64;                // t_k (pre-aggregation)
    float* bufH = ws;  ws += (size_t)N_NODES * 64;                // h_k (aggregated, pre-relu)
    float* sums = ws;  ws += NUM_GRAPHS * 64;
    float* cnt  = ws;

    const int TB = 256;
    dim3 blk(TB);
    int nodeBlocks = (N_NODES + TB - 1) / TB;
    int edgeBlocks = (N_EDGES + TB - 1) / TB;

    // 1) symmetric normalization: deg (with self-loop) -> dinv
    init_deg_kernel<<<nodeBlocks, blk, 0, stream>>>(dinv, N_NODES);
    accum_deg_kernel<<<edgeBlocks, blk, 0, stream>>>(ei, dinv, N_EDGES);
    deg_to_dinv_kernel<<<nodeBlocks, blk, 0, stream>>>(dinv, N_NODES);

    // ---- layer 1: x(128) -> 16 ----
    {
        int waves = MTILES * (16 / 16);
        gemm_wmma_kernel<128, 16, false><<<(waves + 7) / 8, blk, 0, stream>>>(x, W1, bufT, MTILES);
        init_out_kernel<16><<<(N_NODES * 16 + TB - 1) / TB, blk, 0, stream>>>(bufT, dinv, b1, bufH, N_NODES);
        scatter_edges_kernel<16><<<dim3(edgeBlocks, 16 / 4), blk, 0, stream>>>(bufT, ei, dinv, bufH, N_EDGES);
    }
    // ---- layer 2: 16 -> 32 (relu of h1 fused into GEMM A-load) ----
    {
        int waves = MTILES * (32 / 16);
        gemm_wmma_kernel<16, 32, true><<<(waves + 7) / 8, blk, 0, stream>>>(bufH, W2, bufT, MTILES);
        init_out_kernel<32><<<(N_NODES * 32 + TB - 1) / TB, blk, 0, stream>>>(bufT, dinv, b2, bufH, N_NODES);
        scatter_edges_kernel<32><<<dim3(edgeBlocks, 32 / 4), blk, 0, stream>>>(bufT, ei, dinv, bufH, N_EDGES);
    }
    // ---- layer 3: 32 -> 64 ----
    {
        int waves = MTILES * (64 / 16);
        gemm_wmma_kernel<32, 64, true><<<(waves + 7) / 8, blk, 0, stream>>>(bufH, W3, bufT, MTILES);
        init_out_kernel<64><<<(N_NODES * 64 + TB - 1) / TB, blk, 0, stream>>>(bufT, dinv, b3, bufH, N_NODES);
        scatter_edges_kernel<64><<<dim3(edgeBlocks, 64 / 4), blk, 0, stream>>>(bufT, ei, dinv, bufH, N_EDGES);
    }

    // ---- pool + head (relu of h3 fused into pooling) ----
    zero_pool_kernel<<<(NUM_GRAPHS * 64 + TB - 1) / TB, blk, 0, stream>>>(sums, cnt);
    pool_accum_kernel<<<(N_NODES * 64 + TB - 1) / TB, blk, 0, stream>>>(bufH, batch, sums, cnt, N_NODES);
    head_kernel<<<NUM_GRAPHS, 64, 0, stream>>>(sums, cnt, Wl1, bl1, Wl2, bl2, out);
}